// TorchNeighborList_57312043598433
// MI455X (gfx1250) — compile-verified
//
#include <hip/hip_runtime.h>
#include <hip/hip_bf16.h>
#include <stdint.h>

// Neighbor-list for MI455X (gfx1250, wave32).
// Distance phase uses V_WMMA_F32_16X16X4_F32: Gram matrix (P+s)·P^T per 16x16
// tile (K=3 padded to 4), d^2 = |p_i+s|^2 + |p_j|^2 - 2G.  Sort phase is a
// stable counting sort over keys in [0, N] (sentinel N for invalid pairs).

#define CUTOFF_SQ 25.0f
#define MAXN      1024
#define NKEYS     (MAXN + 1)
#define SORT_CHUNK 8192

typedef float v2f __attribute__((ext_vector_type(2)));
typedef float v8f __attribute__((ext_vector_type(8)));

// ---------------------------------------------------------------------------
// Distance kernels: one wave = one 16x16 tile of the NxN distance matrix.
// A-operand layout (16x4 f32): lane L -> row m=L&15, K pair (L>>4)*2.
// C/D layout (16x16 f32): VGPR r, lanes 0-15 -> M=r, lanes 16-31 -> M=8+r.
// ---------------------------------------------------------------------------

__global__ void nl_dist_shifted(const float* __restrict__ pos,
                                const float* __restrict__ box,
                                const int*   __restrict__ shifts,
                                uint8_t*     __restrict__ masks,
                                int N, int S, int Pc)
{
    const int lane = threadIdx.x & 31;
    const int wave = threadIdx.x >> 5;
    const int TD   = N >> 4;                              // tiles per dim
    const long long tilesPerShift = (long long)TD * TD;
    const long long tile = (long long)blockIdx.x * (blockDim.x >> 5) + wave;
    if (tile >= (long long)S * tilesPerShift) return;     // wave-uniform

    const int s  = (int)(tile / tilesPerShift);
    const int tr = (int)(tile % tilesPerShift);
    const int ti = tr / TD;
    const int tj = tr % TD;

    const int sx = shifts[s*3+0], sy = shifts[s*3+1], sz = shifts[s*3+2];
    const float sv0 = sx*box[0] + sy*box[3] + sz*box[6];
    const float sv1 = sx*box[1] + sy*box[4] + sz*box[7];
    const float sv2 = sx*box[2] + sy*box[5] + sz*box[8];

    const int  m16 = lane & 15;
    const bool hi  = (lane >= 16);                        // carries K=2 (+pad K=3)
    const float* pa = pos + (ti*16 + m16)*3;              // A rows: p_i + s
    const float* pb = pos + (tj*16 + m16)*3;              // B cols: p_j
    v2f a, b;
    a.x = hi ? (pa[2] + sv2) : (pa[0] + sv0);
    a.y = hi ? 0.0f          : (pa[1] + sv1);
    b.x = hi ? pb[2] : pb[0];
    b.y = hi ? 0.0f  : pb[1];

    v8f c = {};
    c = __builtin_amdgcn_wmma_f32_16x16x4_f32(false, a, false, b,
                                              (short)0, c, false, false);

    // squared norms, recovered from operand registers via cross-lane shuffles
    const float sa = a.x*a.x + a.y*a.y;
    const float sb = b.x*b.x + b.y*b.y;
    const float nJ = __shfl(sb, m16, 32) + __shfl(sb, m16 + 16, 32);
    const int   mbase = hi ? 8 : 0;
    const long long qbase = (long long)Pc + (long long)s * N * N;
    const int j = tj*16 + m16;
#pragma unroll
    for (int r = 0; r < 8; ++r) {
        const int   m  = mbase + r;
        const float nI = __shfl(sa, m, 32) + __shfl(sa, m + 16, 32);
        const float d2 = nI + nJ - 2.0f * c[r];
        const int   i  = ti*16 + m;
        masks[qbase + (long long)i * N + j] = (d2 < CUTOFF_SQ) ? (uint8_t)1 : (uint8_t)0;
    }
}

// Center cell (zero shift), upper-triangular tiles only; writes masks for
// i<j pairs into triangular packing and records i per pair for fast decode.
__global__ void nl_dist_center(const float* __restrict__ pos,
                               uint8_t*  __restrict__ masks,
                               uint16_t* __restrict__ pairI,
                               int N)
{
    const int lane = threadIdx.x & 31;
    const int wave = threadIdx.x >> 5;
    const int T    = N >> 4;
    const int nTiles = T * (T + 1) / 2;
    const int tile = blockIdx.x * (blockDim.x >> 5) + wave;
    if (tile >= nTiles) return;                           // wave-uniform

    // invert triangular tile index: rowstart(ti) = ti*T - ti*(ti-1)/2
    const float tt = (float)(2*T + 1);
    int ti = (int)((tt - sqrtf(tt*tt - 8.0f*(float)tile)) * 0.5f);
    if (ti >= T) ti = T - 1;
    if (ti < 0)  ti = 0;
    while (ti > 0 && (ti*T - ti*(ti-1)/2) > tile) --ti;
    while (((ti+1)*T - (ti+1)*ti/2) <= tile) ++ti;
    const int tj = ti + (tile - (ti*T - ti*(ti-1)/2));

    const int  m16 = lane & 15;
    const bool hi  = (lane >= 16);
    const float* pa = pos + (ti*16 + m16)*3;
    const float* pb = pos + (tj*16 + m16)*3;
    v2f a, b;
    a.x = hi ? pa[2] : pa[0];
    a.y = hi ? 0.0f  : pa[1];
    b.x = hi ? pb[2] : pb[0];
    b.y = hi ? 0.0f  : pb[1];

    v8f c = {};
    c = __builtin_amdgcn_wmma_f32_16x16x4_f32(false, a, false, b,
                                              (short)0, c, false, false);

    const float sa = a.x*a.x + a.y*a.y;
    const float sb = b.x*b.x + b.y*b.y;
    const float nJ = __shfl(sb, m16, 32) + __shfl(sb, m16 + 16, 32);
    const int   mbase = hi ? 8 : 0;
    const int j = tj*16 + m16;
#pragma unroll
    for (int r = 0; r < 8; ++r) {
        const int   m  = mbase + r;
        const float nI = __shfl(sa, m, 32) + __shfl(sa, m + 16, 32);
        const float d2 = nI + nJ - 2.0f * c[r];
        const int   i  = ti*16 + m;
        if (i < j) {
            const int p = i*N - (i*(i+1) >> 1) + (j - i - 1);  // triu linear index
            masks[p] = (d2 < CUTOFF_SQ) ? (uint8_t)1 : (uint8_t)0;
            pairI[p] = (uint16_t)i;
        }
    }
}

// ---------------------------------------------------------------------------
// Element decode: bidirectional index b in [0, 2P) -> (key, idx_i, idx_j,
// shift ints, sign, valid).  First half uses -shift, second half +shift.
// ---------------------------------------------------------------------------
__device__ __forceinline__ void nl_decode(long long b, long long P, int Pc, int N,
    const uint8_t*  __restrict__ masks,
    const uint16_t* __restrict__ pairI,
    const int*      __restrict__ shifts,
    int& key, int& ii, int& jj, int& sx, int& sy, int& sz, float& sgn, int& valid)
{
    const bool half = (b >= P);
    const int  p = (int)(half ? (b - P) : b);             // P < 2^31
    int i, j;
    sx = 0; sy = 0; sz = 0;
    if (p < Pc) {
        i = (int)pairI[p];
        const int rs = i*N - (i*(i+1) >> 1);
        j = p - rs + i + 1;
    } else {
        const int q  = p - Pc;
        const int NN = N * N;
        const int s  = q / NN;
        const int r  = q - s*NN;
        i = r / N;
        j = r - i*N;
        sx = shifts[s*3+0]; sy = shifts[s*3+1]; sz = shifts[s*3+2];
    }
    valid = (int)masks[p];
    ii  = half ? j : i;
    jj  = half ? i : j;
    sgn = half ? 1.0f : -1.0f;
    key = valid ? ii : N;                                  // sentinel N if invalid
}

// ---- Pass A: per-block key histograms --------------------------------------
__global__ void nl_hist(const uint8_t*  __restrict__ masks,
                        const uint16_t* __restrict__ pairI,
                        const int*      __restrict__ shifts,
                        int* __restrict__ blockHist,
                        long long TOT, long long P, int Pc, int N)
{
    __shared__ int hist[NKEYS];
    for (int k = threadIdx.x; k < NKEYS; k += blockDim.x) hist[k] = 0;
    __syncthreads();
    const long long base = (long long)blockIdx.x * SORT_CHUNK;
    for (int t = threadIdx.x; t < SORT_CHUNK; t += blockDim.x) {
        const long long b = base + t;
        if (b < TOT) {
            int key, ii, jj, sx, sy, sz, valid; float sgn;
            nl_decode(b, P, Pc, N, masks, pairI, shifts,
                      key, ii, jj, sx, sy, sz, sgn, valid);
            atomicAdd(&hist[key], 1);
        }
    }
    __syncthreads();
    int* out = blockHist + (long long)blockIdx.x * NKEYS;
    for (int k = threadIdx.x; k < NKEYS; k += blockDim.x) out[k] = hist[k];
}

// ---- Pass B1: per-key exclusive scan over blocks (one block per key) -------
__global__ void nl_scan_blocks(int* __restrict__ blockHist,
                               int* __restrict__ keyTotal, int B)
{
    __shared__ int vals[256];
    const int k   = blockIdx.x;
    const int tid = threadIdx.x;
    int carry = 0;
    for (int base = 0; base < B; base += blockDim.x) {
        const int idx = base + tid;
        const int v = (idx < B) ? blockHist[(long long)idx * NKEYS + k] : 0;
        vals[tid] = v;
        __syncthreads();
        for (int off = 1; off < blockDim.x; off <<= 1) {
            const int t = (tid >= off) ? vals[tid - off] : 0;
            __syncthreads();
            vals[tid] += t;
            __syncthreads();
        }
        const int incl = vals[tid];
        if (idx < B) blockHist[(long long)idx * NKEYS + k] = carry + (incl - v);
        const int chunkTotal = vals[blockDim.x - 1];
        __syncthreads();
        carry += chunkTotal;
    }
    if (tid == 0) keyTotal[k] = carry;
}

// ---- Pass B2: exclusive scan over keys -------------------------------------
__global__ void nl_scan_keys(const int* __restrict__ keyTotal,
                             int* __restrict__ keyOffset)
{
    if (threadIdx.x == 0 && blockIdx.x == 0) {
        int acc = 0;
        for (int k = 0; k < NKEYS; ++k) { keyOffset[k] = acc; acc += keyTotal[k]; }
    }
}

// ---- Pass C: stable scatter (one wave per block; lane order = b order) -----
__global__ void nl_scatter(const uint8_t*  __restrict__ masks,
                           const uint16_t* __restrict__ pairI,
                           const int*      __restrict__ shifts,
                           const float*    __restrict__ box,
                           const int*      __restrict__ blockHist,
                           const int*      __restrict__ keyOffset,
                           float* __restrict__ out,
                           long long TOT, long long P, int Pc, int N)
{
    __shared__ int cnt[NKEYS];
    __shared__ int pref[NKEYS];
    const int lane = threadIdx.x;                         // blockDim.x == 32
    const int* bh = blockHist + (long long)blockIdx.x * NKEYS;
    for (int k = lane; k < NKEYS; k += 32) {
        cnt[k]  = 0;
        pref[k] = keyOffset[k] + bh[k];
    }
    __syncthreads();

    const float b00 = box[0], b01 = box[1], b02 = box[2];
    const float b10 = box[3], b11 = box[4], b12 = box[5];
    const float b20 = box[6], b21 = box[7], b22 = box[8];

    const long long base = (long long)blockIdx.x * SORT_CHUNK;
    for (int t = 0; t < SORT_CHUNK; t += 32) {
        const long long b = base + t + lane;
        const bool active = (b < TOT);
        int key = -1, ii = 0, jj = 0, sx = 0, sy = 0, sz = 0, valid = 0;
        float sgn = 0.0f;
        if (active)
            nl_decode(b, P, Pc, N, masks, pairI, shifts,
                      key, ii, jj, sx, sy, sz, sgn, valid);

        // stable intra-wave rank among equal keys (inactive lanes key=-1)
        int rank = 0, count = 0, first = -1;
#pragma unroll 4
        for (int l = 0; l < 32; ++l) {
            const int kl = __shfl(key, l, 32);
            if (kl == key) {
                if (l < lane) ++rank;
                ++count;
                if (first < 0) first = l;
            }
        }
        int base0 = 0;
        if (active && rank == 0) base0 = atomicAdd(&cnt[key], count);
        base0 = __shfl(base0, first, 32);

        if (active) {
            const long long pos = (long long)pref[key] + base0 + rank;
            const float vf = valid ? 1.0f : 0.0f;
            out[pos]          = (float)ii;                 // idx_i
            out[TOT + pos]    = (float)jj;                 // idx_j
            const float ox = sgn * (sx*b00 + sy*b10 + sz*b20) * vf;
            const float oy = sgn * (sx*b01 + sy*b11 + sz*b21) * vf;
            const float oz = sgn * (sx*b02 + sy*b12 + sz*b22) * vf;
            const long long ob = 2*TOT + pos*3;
            out[ob + 0] = ox; out[ob + 1] = oy; out[ob + 2] = oz;  // offset
            out[5*TOT + pos]  = vf;                        // valid
        }
    }
}

// ---------------------------------------------------------------------------
extern "C" void kernel_launch(void* const* d_in, const int* in_sizes, int n_in,
                              void* d_out, int out_size, void* d_ws, size_t ws_size,
                              hipStream_t stream)
{
    (void)n_in; (void)out_size; (void)ws_size;
    const float* pos    = (const float*)d_in[0];   // [N,3] f32
    const float* box    = (const float*)d_in[1];   // [3,3] f32
    const int*   shifts = (const int*)d_in[2];     // [S,3] i32

    const int N  = in_sizes[0] / 3;                // 1024
    const int S  = in_sizes[2] / 3;                // 13
    const int Pc = N * (N - 1) / 2;
    const long long P   = (long long)Pc + (long long)S * N * N;
    const long long TOT = 2 * P;

    // workspace layout
    char* ws = (char*)d_ws;
    size_t off = 0;
    uint8_t* masks = (uint8_t*)(ws + off);
    off += (size_t)P;            off = (off + 255) & ~(size_t)255;
    uint16_t* pairI = (uint16_t*)(ws + off);
    off += (size_t)Pc * 2;       off = (off + 255) & ~(size_t)255;
    const int B = (int)((TOT + SORT_CHUNK - 1) / SORT_CHUNK);
    int* blockHist = (int*)(ws + off);
    off += (size_t)B * NKEYS * 4; off = (off + 255) & ~(size_t)255;
    int* keyTotal = (int*)(ws + off);
    off += (size_t)NKEYS * 4;     off = (off + 255) & ~(size_t)255;
    int* keyOffset = (int*)(ws + off);

    // ---- phase 1: distances / masks via WMMA -----------------------------
    {
        const int TD = N >> 4;
        const long long tiles = (long long)S * TD * TD;        // 53248
        const int wavesPerBlock = 8;                           // 256 threads
        const int blocks = (int)((tiles + wavesPerBlock - 1) / wavesPerBlock);
        nl_dist_shifted<<<blocks, 256, 0, stream>>>(pos, box, shifts, masks, N, S, Pc);

        const int T = N >> 4;
        const int cTiles = T * (T + 1) / 2;                    // 2080
        const int cBlocks = (cTiles + wavesPerBlock - 1) / wavesPerBlock;
        nl_dist_center<<<cBlocks, 256, 0, stream>>>(pos, masks, pairI, N);
    }

    // ---- phase 2: stable counting sort + scatter -------------------------
    nl_hist<<<B, 256, 0, stream>>>(masks, pairI, shifts, blockHist, TOT, P, Pc, N);
    nl_scan_blocks<<<NKEYS, 256, 0, stream>>>(blockHist, keyTotal, B);
    nl_scan_keys<<<1, 1, 0, stream>>>(keyTotal, keyOffset);
    nl_scatter<<<B, 32, 0, stream>>>(masks, pairI, shifts, box, blockHist,
                                     keyOffset, (float*)d_out, TOT, P, Pc, N);
}